// CNN_Comp_29240137351522
// MI455X (gfx1250) — compile-verified
//
#include <hip/hip_runtime.h>

typedef __attribute__((ext_vector_type(2))) float v2f;
typedef __attribute__((ext_vector_type(4))) float v4f;
typedef __attribute__((ext_vector_type(8))) float v8f;

#define N_FFT   4096
#define CLASSN  2048
#define TWO_PI  6.28318530717958647692f
// LDS skew: +1 float every 16 -> stride-16 accesses become stride-17 (conflict-free)
__device__ __forceinline__ int pad(int i) { return i + (i >> 4); }
#define LDS_FLOATS (N_FFT + (N_FFT >> 4))   // 4352

// ---------------------------------------------------------------------------
// Setup: C[k] = (FFT4096(w0)[k])^2 * FFT4096(wl)[k] / 4096   (row-independent)
// ---------------------------------------------------------------------------
__global__ __launch_bounds__(128) void build_spectrum_kernel(
    const float* __restrict__ w0r, const float* __restrict__ w0i,
    const float* __restrict__ wlr, const float* __restrict__ wli,
    float2* __restrict__ C) {
  int k = blockIdx.x * blockDim.x + threadIdx.x;
  if (k >= N_FFT) return;
  float ar = 0.f, ai = 0.f;
  for (int n = 0; n < 129; ++n) {
    int t = (k * n) & (N_FFT - 1);                 // exact phase reduction
    float s, c; __sincosf(-(TWO_PI / N_FFT) * (float)t, &s, &c);
    ar += w0r[n] * c - w0i[n] * s;
    ai += w0r[n] * s + w0i[n] * c;
  }
  float br = 0.f, bi = 0.f;
  for (int n = 0; n < 257; ++n) {
    int t = (k * n) & (N_FFT - 1);
    float s, c; __sincosf(-(TWO_PI / N_FFT) * (float)t, &s, &c);
    br += wlr[n] * c - wli[n] * s;
    bi += wlr[n] * s + wli[n] * c;
  }
  float a2r = ar * ar - ai * ai;
  float a2i = 2.f * ar * ai;
  float2 o;
  o.x = (a2r * br - a2i * bi) * (1.0f / N_FFT);    // fold IFFT 1/N scale
  o.y = (a2r * bi + a2i * br) * (1.0f / N_FFT);
  C[k] = o;
}

// ---------------------------------------------------------------------------
// WMMA helper: D = A(16x4 f32) * B(4x16 f32) + C
// ---------------------------------------------------------------------------
__device__ __forceinline__ v8f wmma4(v2f a, v2f b, v8f c) {
  return __builtin_amdgcn_wmma_f32_16x16x4_f32(
      false, a, false, b, (short)0, c, false, false);
}

// DFT-16 matrix in WMMA B layout (slice s, .x -> row 4s+2hi, .y -> row +1).
struct B16 { v2f re[4], im[4], imn[4]; };

// ---------------------------------------------------------------------------
// One radix-16 stage (compile-time specialized), in-place per 16x16 plane.
// Index digits: in n = n1 + 16*m1 + 256*m2 ; out k = j2 + 16*j1 + 256*k1.
//   STAGE 0: contract m2 (A[M=m1][K=m2]),  post-twiddle W256^(m1*j2)
//   STAGE 1: contract m1 (A[M=j2][K=m1]),  post-twiddle W4096^(n1*(j2+16*j1))
//   STAGE 2: contract n1 (A[M=j2][K=n1]),  no twiddle
// ---------------------------------------------------------------------------
template <int STAGE, bool INV>
__device__ __forceinline__ void dft_stage(float* __restrict__ sre,
                                          float* __restrict__ sim,
                                          int wave, int hi, int lm,
                                          const B16& B) {
  const float sgn = INV ? 1.0f : -1.0f;

  // Stage-0 twiddles depend only on (M,N) -> hoist out of the plane loop.
  float t0c[8], t0s[8];
  if (STAGE == 0) {
#pragma unroll
    for (int v = 0; v < 8; ++v) {
      int M = v + 8 * hi;                     // angle < 2*pi (M*lm <= 225)
      __sincosf(sgn * (TWO_PI / 256.0f) * (float)(M * lm), &t0s[v], &t0c[v]);
    }
  }

#pragma unroll 1                              // rolled: keeps VGPRs < 256
  for (int pi = 0; pi < 4; ++pi) {
    const int p = wave + 4 * pi;

    // Load complex A tile (whole 16x16 plane into the wave's registers).
    v2f are[4], aim[4];
#pragma unroll
    for (int s = 0; s < 4; ++s) {
      int k0 = 4 * s + 2 * hi;
      int i0, i1;
      if (STAGE == 0)      { i0 = p + 16 * lm + 256 * k0; i1 = i0 + 256; }
      else if (STAGE == 1) { i0 = p + 16 * k0 + 256 * lm; i1 = i0 + 16;  }
      else                 { i0 = k0 + 16 * p + 256 * lm; i1 = i0 + 1;   }
      are[s].x = sre[pad(i0)]; are[s].y = sre[pad(i1)];
      aim[s].x = sim[pad(i0)]; aim[s].y = sim[pad(i1)];
    }

    // Complex 16x16x16 matmul: 16 x v_wmma_f32_16x16x4_f32
    v8f cre = {}; v8f cim = {};
#pragma unroll
    for (int s = 0; s < 4; ++s) {
      cre = wmma4(are[s], B.re[s],  cre);
      cre = wmma4(aim[s], B.imn[s], cre);     // - Aim*Bim
      cim = wmma4(are[s], B.im[s],  cim);
      cim = wmma4(aim[s], B.re[s],  cim);
    }

    // Twiddle + in-place store. D layout: M = v + 8*hi, N = lane&15.
#pragma unroll
    for (int v = 0; v < 8; ++v) {
      int M = v + 8 * hi, Nn = lm;
      float xr = cre[v], xi = cim[v];
      if (STAGE == 0) {
        float yr = xr * t0c[v] - xi * t0s[v];
        float yi = xr * t0s[v] + xi * t0c[v];
        xr = yr; xi = yi;
      } else if (STAGE == 1) {
        float ts, tc;                          // p*(M+16N) <= 3825 < 4096
        __sincosf(sgn * (TWO_PI / N_FFT) * (float)(p * (M + 16 * Nn)),
                  &ts, &tc);
        float yr = xr * tc - xi * ts;
        float yi = xr * ts + xi * tc;
        xr = yr; xi = yi;
      }
      int wi;
      if (STAGE == 0)      wi = p + 16 * M + 256 * Nn;
      else if (STAGE == 1) wi = p + 16 * Nn + 256 * M;
      else                 wi = M + 16 * p + 256 * Nn;
      sre[pad(wi)] = xr; sim[pad(wi)] = xi;
    }
  }
  __syncthreads();
}

// ---------------------------------------------------------------------------
// In-LDS 4096-pt FFT (INV=false) / unscaled IFFT (INV=true), radix-16^3.
// 128 threads = 4 wave32; each wave owns 4 of the 16 planes per stage.
// ---------------------------------------------------------------------------
template <bool INV>
__device__ void fft_pass(float* __restrict__ sre, float* __restrict__ sim) {
  const int tid  = threadIdx.x;
  const int wave = tid >> 5;
  const int lane = tid & 31;
  const int hi   = (lane >> 4) & 1;
  const int lm   = lane & 15;
  const float sgn = INV ? 1.0f : -1.0f;

  B16 B;
#pragma unroll
  for (int s = 0; s < 4; ++s) {
    int k0 = 4 * s + 2 * hi;
    float s0, c0, s1, c1;
    __sincosf(sgn * (TWO_PI / 16.0f) * (float)((k0 * lm) & 15), &s0, &c0);
    __sincosf(sgn * (TWO_PI / 16.0f) * (float)(((k0 + 1) * lm) & 15), &s1, &c1);
    B.re[s].x = c0;  B.re[s].y = c1;
    B.im[s].x = s0;  B.im[s].y = s1;
    B.imn[s].x = -s0; B.imn[s].y = -s1;
  }

  dft_stage<0, INV>(sre, sim, wave, hi, lm, B);
  dft_stage<1, INV>(sre, sim, wave, hi, lm, B);
  dft_stage<2, INV>(sre, sim, wave, hi, lm, B);
}

// ---------------------------------------------------------------------------
// Per-row: load+pad -> FFT -> X^2 * C -> IFFT -> |center 2048|
// ---------------------------------------------------------------------------
__global__ __launch_bounds__(128) void fft_conv_abs_kernel(
    const float* __restrict__ xr, const float* __restrict__ xi,
    const float2* __restrict__ C, float* __restrict__ out) {
  __shared__ float sre[LDS_FLOATS];
  __shared__ float sim[LDS_FLOATS];
  const int row = blockIdx.x;
  const int tid = threadIdx.x;

  // Vectorized, non-temporal row load (b128 transactions; data used once).
  // i0 = 4*t is 4-aligned, so pad(i0)..pad(i0)+3 stay contiguous in LDS.
  const v4f* xr4 = (const v4f*)(xr + (size_t)row * 1024);
  const v4f* xi4 = (const v4f*)(xi + (size_t)row * 1024);
  for (int t = tid; t < 256; t += 128) {     // 256 x v4f = 1024 floats
    v4f vr = __builtin_nontemporal_load(&xr4[t]);
    v4f vi = __builtin_nontemporal_load(&xi4[t]);
    int a = pad(4 * t);
    sre[a + 0] = vr.x; sre[a + 1] = vr.y; sre[a + 2] = vr.z; sre[a + 3] = vr.w;
    sim[a + 0] = vi.x; sim[a + 1] = vi.y; sim[a + 2] = vi.z; sim[a + 3] = vi.w;
  }
  for (int i = 1024 + tid; i < N_FFT; i += 128) {   // zero-pad tail
    sre[pad(i)] = 0.f; sim[pad(i)] = 0.f;
  }
  __syncthreads();

  fft_pass<false>(sre, sim);                 // forward FFT (barriers inside)

  for (int k = tid; k < N_FFT; k += 128) {   // Y = X^2 * C
    int a = pad(k);
    float r = sre[a], m = sim[a];
    float s2r = r * r - m * m;
    float s2i = 2.f * r * m;
    float2 c = C[k];
    sre[a] = s2r * c.x - s2i * c.y;
    sim[a] = s2r * c.y + s2i * c.x;
  }
  __syncthreads();

  fft_pass<true>(sre, sim);                  // inverse FFT (1/N folded in C)

  // Crop offset 255, magnitude; vectorized non-temporal b128 output stores.
  v4f* out4 = (v4f*)(out + (size_t)row * CLASSN);
  for (int t = tid; t < CLASSN / 4; t += 128) {
    int i0 = 255 + 4 * t;
    v4f o;
    { float r = sre[pad(i0 + 0)], m = sim[pad(i0 + 0)]; o.x = sqrtf(r*r + m*m); }
    { float r = sre[pad(i0 + 1)], m = sim[pad(i0 + 1)]; o.y = sqrtf(r*r + m*m); }
    { float r = sre[pad(i0 + 2)], m = sim[pad(i0 + 2)]; o.z = sqrtf(r*r + m*m); }
    { float r = sre[pad(i0 + 3)], m = sim[pad(i0 + 3)]; o.w = sqrtf(r*r + m*m); }
    __builtin_nontemporal_store(o, &out4[t]);
  }
}

// ---------------------------------------------------------------------------
extern "C" void kernel_launch(void* const* d_in, const int* in_sizes, int n_in,
                              void* d_out, int out_size, void* d_ws, size_t ws_size,
                              hipStream_t stream) {
  (void)in_sizes; (void)n_in; (void)out_size; (void)ws_size;
  const float* x_real  = (const float*)d_in[0];
  const float* x_imag  = (const float*)d_in[1];
  const float* w0_real = (const float*)d_in[2];
  const float* w0_imag = (const float*)d_in[3];
  const float* wl_real = (const float*)d_in[4];
  const float* wl_imag = (const float*)d_in[5];
  float*  out = (float*)d_out;
  float2* C   = (float2*)d_ws;               // 4096 * 8 B = 32 KB scratch

  build_spectrum_kernel<<<dim3(N_FFT / 128), dim3(128), 0, stream>>>(
      w0_real, w0_imag, wl_real, wl_imag, C);
  fft_conv_abs_kernel<<<dim3(4096), dim3(128), 0, stream>>>(
      x_real, x_imag, C, out);
}